// TimeGreedyModel_75694503624833
// MI455X (gfx1250) — compile-verified
//
#include <hip/hip_runtime.h>
#include <stdint.h>

// Per-row stable argsort:
//   key = mask ? +inf : time ;  pred = stable_argsort(key) with tail = pad_value
//   valid = popcount(!mask) per row
// One workgroup (256 threads = 8 wave32) per row. Row staged into LDS with
// gfx1250 async global->LDS loads, then an 8-pass 4-bit LSD radix sort fully
// in LDS (stable by construction: bin-major scan, consecutive per-thread
// chunks, in-order in-chunk scatter). Masked entries keyed 0xFFFFFFFF — they
// land past `valid` and are overwritten by pad_value, so their internal order
// is irrelevant (matches the reference's +inf fill).

#define LROW   2048
#define TBLK   256
#define EPT    (LROW / TBLK)   // 8 elements per thread
#define NBIN   16
#define PASSES 8

typedef int v4i __attribute__((__vector_size__(16)));

__device__ __forceinline__ void async_copy_b128(const void* g, void* l) {
#if __has_builtin(__builtin_amdgcn_global_load_async_to_lds_b128)
  // (global src, lds dst, imm offset, imm cpol)
  __builtin_amdgcn_global_load_async_to_lds_b128((v4i*)g, (v4i*)l, 0, 0);
#else
  *(uint4*)l = *(const uint4*)g;  // fallback: synchronous 16B copy
#endif
}

__device__ __forceinline__ void wait_async_lds() {
#if __has_builtin(__builtin_amdgcn_s_wait_asynccnt)
  __builtin_amdgcn_s_wait_asynccnt(0);
#elif __has_builtin(__builtin_amdgcn_global_load_async_to_lds_b128)
  asm volatile("s_wait_asynccnt 0" ::: "memory");
#endif
}

__global__ __launch_bounds__(TBLK) void sort_rows_kernel(
    const float* __restrict__ time, const uint8_t* __restrict__ mask,
    const int* __restrict__ pad, int* __restrict__ pred,
    int* __restrict__ valid_out, int B) {
  __shared__ uint32_t keyA[LROW];
  __shared__ uint32_t keyB[LROW];          // also the staging buffer for time
  __shared__ uint16_t idxA[LROW];
  __shared__ uint16_t idxB[LROW];          // also the staging buffer for mask
  __shared__ uint16_t hist[NBIN * TBLK];   // bin-major: hist[b*TBLK + t]
  __shared__ uint32_t partial[TBLK];
  __shared__ uint32_t vcount;

  const int row = blockIdx.x;
  const int tid = threadIdx.x;
  if (row >= B) return;

  const float*   trow = time + (size_t)row * LROW;
  const uint8_t* mrow = mask + (size_t)row * LROW;

  // ---- async stage: time row (8KB) -> keyB, mask row (2KB) -> idxB ----
  {
    const char* tg = (const char*)trow;
    char*       tl = (char*)keyB;
    async_copy_b128(tg + tid * 16, tl + tid * 16);
    async_copy_b128(tg + 4096 + tid * 16, tl + 4096 + tid * 16);
    if (tid < 128)
      async_copy_b128((const char*)mrow + tid * 16, (char*)idxB + tid * 16);
  }
  if (tid == 0) vcount = 0;
  wait_async_lds();
  __syncthreads();

  // ---- build sortable keys + count valid ----
  {
    const uint8_t* mstage = (const uint8_t*)idxB;
    uint32_t lv = 0;
    for (int e = 0; e < EPT; ++e) {
      int i = tid + e * TBLK;
      uint32_t bits = keyB[i];
      // monotonic float -> uint mapping (handles negatives generically)
      uint32_t mk = (bits & 0x80000000u) ? ~bits : (bits | 0x80000000u);
      uint8_t m = mstage[i];
      keyA[i] = m ? 0xFFFFFFFFu : mk;
      idxA[i] = (uint16_t)i;
      lv += (m == 0);
    }
    atomicAdd(&vcount, lv);
  }
  __syncthreads();

  // ---- 8 passes of stable 4-bit LSD radix sort in LDS ----
  for (int p = 0; p < PASSES; ++p) {
    const int shift = p * 4;
    uint32_t* sk = (p & 1) ? keyB : keyA;
    uint32_t* dk = (p & 1) ? keyA : keyB;
    uint16_t* si = (p & 1) ? idxB : idxA;
    uint16_t* di = (p & 1) ? idxA : idxB;

    for (int i = tid; i < NBIN * TBLK; i += TBLK) hist[i] = 0;
    __syncthreads();

    const int base = tid * EPT;  // consecutive chunk -> stability
    for (int e = 0; e < EPT; ++e) {
      uint32_t d = (sk[base + e] >> shift) & (NBIN - 1);
      hist[d * TBLK + tid] += 1;  // thread-private column, no race
    }
    __syncthreads();

    // exclusive scan over hist[0 .. NBIN*TBLK) in bin-major order
    {
      uint32_t run = 0;
      const int s0 = tid * NBIN;  // NBIN*TBLK / TBLK == NBIN entries each
      for (int i = 0; i < NBIN; ++i) {
        uint32_t v = hist[s0 + i];
        hist[s0 + i] = (uint16_t)run;
        run += v;
      }
      partial[tid] = run;
      __syncthreads();
      for (int off = 1; off < TBLK; off <<= 1) {
        uint32_t add = (tid >= off) ? partial[tid - off] : 0;
        __syncthreads();
        partial[tid] += add;
        __syncthreads();
      }
      uint32_t excl = tid ? partial[tid - 1] : 0;
      for (int i = 0; i < NBIN; ++i) hist[s0 + i] += (uint16_t)excl;
    }
    __syncthreads();

    // stable scatter
    for (int e = 0; e < EPT; ++e) {
      uint32_t k  = sk[base + e];
      uint16_t id = si[base + e];
      uint32_t d  = (k >> shift) & (NBIN - 1);
      uint32_t pos = hist[d * TBLK + tid];
      hist[d * TBLK + tid] = (uint16_t)(pos + 1);
      dk[pos] = k;
      di[pos] = id;
    }
    __syncthreads();
  }
  // PASSES even -> final data resides in keyA/idxA

  // ---- emit pred row + valid count ----
  const int padv   = pad[0];
  const int validv = (int)vcount;
  int* prow = pred + (size_t)row * LROW;
  for (int e = 0; e < EPT; ++e) {
    int posi = tid + e * TBLK;
    int ordv = (int)idxA[posi];
    prow[posi] = (posi < validv) ? ordv : padv;
  }
  if (tid == 0) valid_out[row] = validv;
}

extern "C" void kernel_launch(void* const* d_in, const int* in_sizes, int n_in,
                              void* d_out, int out_size, void* d_ws,
                              size_t ws_size, hipStream_t stream) {
  (void)n_in; (void)out_size; (void)d_ws; (void)ws_size;
  const float*   time = (const float*)d_in[0];
  const uint8_t* mask = (const uint8_t*)d_in[1];  // jnp.bool_ -> 1 byte/elem
  const int*     pad  = (const int*)d_in[2];
  const int B = in_sizes[0] / LROW;
  int* pred  = (int*)d_out;
  int* valid = pred + (size_t)B * LROW;
  sort_rows_kernel<<<B, TBLK, 0, stream>>>(time, mask, pad, pred, valid, B);
}